// SeparatedGNNSystemV3_15109694948037
// MI455X (gfx1250) — compile-verified
//
#include <hip/hip_runtime.h>
#include <stdint.h>
#include <stddef.h>

// ---------------------------------------------------------------------------
// SeparatedGNNSystem gather: out[i] = is_last(i) ? solvent[map[i]]
//                                                : monomer[i - map[i]]
// Pure streaming copy (0 FLOPs, ~0.82 GB traffic -> ~35us at 23.3 TB/s floor).
// CDNA5 async global<->LDS data-mover path (ASYNCcnt): 128b/lane async
// gathers into a 64KB LDS staging tile, s_wait_asynccnt, 128b/lane async
// stores to the contiguous output. Payload never touches VGPRs.
// The per-wave row index is wave-uniform by construction (256 threads, 64
// chunks/row => t>>6 constant within a wave32), enforced via readfirstlane so
// the mapping reads + boundary select lower to scalar (SMEM/SALU) code.
// ---------------------------------------------------------------------------

#define AS1 __attribute__((address_space(1)))
#define AS3 __attribute__((address_space(3)))

typedef int v4i __attribute__((ext_vector_type(4)));

#ifndef __has_builtin
#define __has_builtin(x) 0
#endif

__device__ __forceinline__ void async_gather_b128(const void* gsrc, void* lds_dst) {
#if __has_builtin(__builtin_amdgcn_global_load_async_to_lds_b128)
  __builtin_amdgcn_global_load_async_to_lds_b128(
      (AS1 v4i*)gsrc, (AS3 v4i*)lds_dst, /*imm offset*/0, /*cpol*/0);
#else
  unsigned loff = (unsigned)(size_t)(AS3 char*)lds_dst;
  asm volatile("global_load_async_to_lds_b128 %0, %1, off"
               :
               : "v"(loff), "v"(gsrc)
               : "memory");
#endif
}

__device__ __forceinline__ void async_scatter_b128(void* gdst, const void* lds_src) {
#if __has_builtin(__builtin_amdgcn_global_store_async_from_lds_b128)
  __builtin_amdgcn_global_store_async_from_lds_b128(
      (AS1 v4i*)gdst, (AS3 v4i*)lds_src, /*imm offset*/0, /*cpol*/0);
#else
  unsigned loff = (unsigned)(size_t)(AS3 char*)lds_src;
  asm volatile("global_store_async_from_lds_b128 %0, %1, off"
               :
               : "v"(gdst), "v"(loff)
               : "memory");
#endif
}

__device__ __forceinline__ void wait_asynccnt0() {
#if __has_builtin(__builtin_amdgcn_s_wait_asynccnt)
  __builtin_amdgcn_s_wait_asynccnt(0);
#else
  asm volatile("s_wait_asynccnt 0" ::: "memory");
#endif
}

#define ROW_BYTES  1024u               // 256 floats per row
#define CHUNKS_ROW 64u                 // float4 chunks per row
#define BT         256u                // threads per block (8 wave32)
#define CPT        16u                 // chunks per thread
#define TILE       (BT * CPT)          // 4096 chunks = 64 KB LDS tile

__global__ __launch_bounds__(BT)
void polymer_gather_async(const float* __restrict__ mono,
                          const float* __restrict__ solv,
                          const int*   __restrict__ map,
                          float*       __restrict__ out,
                          int nmol_s)
{
  __shared__ float4 stage[TILE];       // 64 KB staging tile

  const unsigned nmol        = (unsigned)nmol_s;
  const unsigned totalChunks = nmol * CHUNKS_ROW;           // 25.6M < 2^31
  const unsigned base        = (unsigned)blockIdx.x * TILE; // chunk index
  const unsigned t           = threadIdx.x;

  const char* monoB = (const char*)mono;
  const char* solvB = (const char*)solv;
  char*       outB  = (char*)out;

  // base % 4096 == 0 and (j*BT) % 64 == 0 => lane offset is loop-invariant.
  const unsigned laneOff = (t & 63u) * 16u;        // byte offset within row

  if (base + TILE < totalChunks) {
    // ---- fast path: full tile AND row+1 < nmol for every row in tile ----
    // 32 lanes of a wave span t-ranges inside one 64-thread half => t>>6 is
    // wave-uniform; readfirstlane makes that visible to the compiler so the
    // map[] reads and boundary select become scalar.
    const unsigned waveRow0 =
        (unsigned)__builtin_amdgcn_readfirstlane((int)((base >> 6) + (t >> 6)));

#pragma unroll
    for (unsigned j = 0; j < CPT; ++j) {
      const unsigned row = waveRow0 + j * (BT / 64u);  // +4 rows per j
      const int p  = map[row];                         // scalar load
      const int pn = map[row + 1u];                    // safe: row+1 < nmol
      const char* srcRow = (pn != p)
          ? (solvB + (unsigned)p * ROW_BYTES)                    // last in group
          : (monoB + (row - (unsigned)p) * ROW_BYTES);           // monomer rank
      async_gather_b128(srcRow + laneOff, &stage[j * BT + t]);
    }

    wait_asynccnt0();   // lane-private LDS slots: ASYNCcnt ordering suffices

    char* outTile = outB + base * 16u;
#pragma unroll
    for (unsigned j = 0; j < CPT; ++j) {
      async_scatter_b128(outTile + (j * BT + t) * 16u, &stage[j * BT + t]);
    }
  } else {
    // ---- tail path: per-chunk guards + clamped boundary read ----
#pragma unroll
    for (unsigned j = 0; j < CPT; ++j) {
      const unsigned c = base + j * BT + t;
      if (c < totalChunks) {
        const unsigned row = c >> 6;
        const unsigned lo  = (c & 63u) * 16u;
        const int p = map[row];
        const unsigned nclamp = (row + 1u < nmol) ? (row + 1u) : (nmol - 1u);
        const bool last = (row + 1u >= nmol) | (map[nclamp] != p);
        const char* src = last ? (solvB + (unsigned)p * ROW_BYTES + lo)
                               : (monoB + (row - (unsigned)p) * ROW_BYTES + lo);
        async_gather_b128(src, &stage[j * BT + t]);
      }
    }
    wait_asynccnt0();
#pragma unroll
    for (unsigned j = 0; j < CPT; ++j) {
      const unsigned c = base + j * BT + t;
      if (c < totalChunks) {
        async_scatter_b128(outB + c * 16u, &stage[j * BT + t]);
      }
    }
  }
  // S_ENDPGM performs an implicit wait-idle for outstanding async stores.
}

extern "C" void kernel_launch(void* const* d_in, const int* in_sizes, int n_in,
                              void* d_out, int out_size, void* d_ws, size_t ws_size,
                              hipStream_t stream) {
  const float* mono = (const float*)d_in[0];   // [P*3, 256] f32
  const float* solv = (const float*)d_in[1];   // [P, 256]   f32
  const int*   map  = (const int*)d_in[2];     // [P*4]      i32
  float*       out  = (float*)d_out;           // [P*4, 256] f32

  const int nmol = in_sizes[2];                // 400,000 molecules
  const unsigned totalChunks = (unsigned)nmol * CHUNKS_ROW;
  const int blocks = (int)((totalChunks + TILE - 1u) / TILE);  // 6250

  polymer_gather_async<<<blocks, BT, 0, stream>>>(mono, solv, map, out, nmol);
}